// Vec2Im_26096221291019
// MI455X (gfx1250) — compile-verified
//
#include <hip/hip_runtime.h>
#include <cstdint>
#include <cstddef>

// Problem constants from the reference
#define IMG_H 512
#define IMG_W 512
#define N_RX  34

// ---------------------------------------------------------------------------
// Kernel 1: stream 256 MB of zeros to d_out using the CDNA5 async
// LDS -> global store DMA path (GLOBAL_STORE_ASYNC_FROM_LDS_B128, ASYNCcnt).
// All lanes source the same 16 zero bytes in LDS (broadcast read); stores are
// non-temporal since the 256 MB output exceeds the 192 MB L2.
// ---------------------------------------------------------------------------
__global__ void fill_zero_async(float* __restrict__ out,
                                unsigned long long n_chunks16) {
    __shared__ __align__(16) unsigned int zbuf[4];
    if (threadIdx.x < 4) zbuf[threadIdx.x] = 0u;
    __syncthreads();

    unsigned int lds_off = (unsigned int)(uintptr_t)(&zbuf[0]);

    unsigned long long tid    = (unsigned long long)blockIdx.x * blockDim.x + threadIdx.x;
    unsigned long long stride = (unsigned long long)gridDim.x * blockDim.x;
    unsigned long long base   = (unsigned long long)(uintptr_t)out;

    for (unsigned long long i = tid; i < n_chunks16; i += stride) {
        unsigned long long gaddr = base + (i << 4);
        asm volatile(
            "global_store_async_from_lds_b128 %0, %1, off th:TH_STORE_NT"
            :
            : "v"(gaddr), "v"(lds_off)
            : "memory");
    }
    // Drain the async store queue before the wave retires.
    asm volatile("s_wait_asynccnt 0" ::: "memory");
}

// ---------------------------------------------------------------------------
// Kernel 2: scatter the 128*34 weighted points. One thread per (b, rx).
// p   = raw*dev_w[rx] + mask*dev_b[rx]
// p   = p*cat_w[cat]  + mask*cat_b[cat]
// out[b,0,yy,xx] = p ; out[b,1,yy,xx] = raw
// ---------------------------------------------------------------------------
__global__ void scatter_points(const float4* __restrict__ xv,
                               const float*  __restrict__ dev_w,
                               const float*  __restrict__ dev_b,
                               const float*  __restrict__ cat_w,
                               const float*  __restrict__ cat_b,
                               float* __restrict__ out,
                               int total) {
    int i = blockIdx.x * blockDim.x + threadIdx.x;
    if (i >= total) return;

    int rx = i % N_RX;
    int b  = i / N_RX;

    float4 v   = xv[i];            // [power, x, y, cat]
    float raw  = v.x;
    float mask = (raw != 0.0f) ? 1.0f : 0.0f;

    float p = fmaf(raw, dev_w[rx], mask * dev_b[rx]);
    int   c = (int)v.w;            // astype(int32) truncation
    p = fmaf(p, cat_w[c], mask * cat_b[c]);

    int xx = __float2int_rn(v.y);  // jnp.round -> round-to-nearest-even
    int yy = __float2int_rn(v.z);

    size_t plane = (size_t)IMG_H * IMG_W;
    size_t base  = (size_t)b * (2ull * plane) + (size_t)yy * IMG_W + (size_t)xx;
    out[base]         = p;     // channel 0
    out[base + plane] = raw;   // channel 1
}

extern "C" void kernel_launch(void* const* d_in, const int* in_sizes, int n_in,
                              void* d_out, int out_size, void* d_ws, size_t ws_size,
                              hipStream_t stream) {
    const float4* xv    = (const float4*)d_in[0];  // (128, 34, 4) f32
    const float*  dev_w = (const float*)d_in[1];   // (34,)
    const float*  dev_b = (const float*)d_in[2];   // (34,)
    const float*  cat_w = (const float*)d_in[3];   // (5,)
    const float*  cat_b = (const float*)d_in[4];   // (5,)
    float* out = (float*)d_out;                    // (128, 2, 512, 512) f32

    // Zero-fill: out_size floats -> out_size/4 16-byte chunks (exactly divisible).
    unsigned long long n_chunks16 = (unsigned long long)out_size / 4ull;
    fill_zero_async<<<8192, 256, 0, stream>>>(out, n_chunks16);

    // Scatter the points on top of the zero background.
    int total = in_sizes[0] / 4;   // 128 * 34 = 4352 records
    scatter_points<<<(total + 255) / 256, 256, 0, stream>>>(
        xv, dev_w, dev_b, cat_w, cat_b, out, total);
}